// Sampler_72000831750737
// MI455X (gfx1250) — compile-verified
//
#include <hip/hip_runtime.h>
#include <hip/hip_bf16.h>
#include <math.h>

// One workgroup per row. 1024 threads = 32 wave32 waves (CDNA5 WGP).
#define BS    1024
#define NW    (BS / 32)
#define NBINS 2048
#define CH    (4 * BS)   // pass-1 chunk: 4096 floats = 16 KB

// ---------------------------------------------------------------------------
// gfx1250 async global->LDS copies (ASYNCcnt-tracked), proven to lower in
// round 2 (global_load_async_to_lds_b32). Guarded with a plain-load fallback.
// ---------------------------------------------------------------------------
#if defined(__HIP_DEVICE_COMPILE__) && __has_builtin(__builtin_amdgcn_global_load_async_to_lds_b32)
#define USE_ASYNC 1
#else
#define USE_ASYNC 0
#endif

#if USE_ASYNC
typedef __attribute__((address_space(1))) int g_int;
typedef __attribute__((address_space(3))) int l_int;
__device__ __forceinline__ void async_copy_b32(const float* g, float* lds) {
  __builtin_amdgcn_global_load_async_to_lds_b32((g_int*)g, (l_int*)lds, 0, 0);
}
__device__ __forceinline__ void wait_async_le4() {   // 1 chunk (4 instr) may remain
#if __has_builtin(__builtin_amdgcn_s_wait_asynccnt)
  __builtin_amdgcn_s_wait_asynccnt(4);
#else
  asm volatile("s_wait_asynccnt 4" ::: "memory");
#endif
}
__device__ __forceinline__ void wait_async_le0() {
#if __has_builtin(__builtin_amdgcn_s_wait_asynccnt)
  __builtin_amdgcn_s_wait_asynccnt(0);
#else
  asm volatile("s_wait_asynccnt 0" ::: "memory");
#endif
}
#endif

// Order-preserving float -> uint key (ascending key == ascending float).
__device__ __forceinline__ unsigned key_of(float f) {
  unsigned u = __float_as_uint(f);
  return (u & 0x80000000u) ? ~u : (u | 0x80000000u);
}

// Deterministic block argmax (tie -> smallest index, matching jnp.argmax).
__device__ __forceinline__ void block_argmax(float v, int idx, int tid,
                                             float* redV, int* redI,
                                             float* outV, int* outI) {
#pragma unroll
  for (int off = 16; off > 0; off >>= 1) {
    float ov = __shfl_down(v, off, 32);
    int   oi = __shfl_down(idx, off, 32);
    if (ov > v || (ov == v && oi < idx)) { v = ov; idx = oi; }
  }
  if ((tid & 31) == 0) { redV[tid >> 5] = v; redI[tid >> 5] = idx; }
  __syncthreads();
  if (tid == 0) {
    float bv = redV[0]; int bi = redI[0];
    for (int w = 1; w < NW; ++w)
      if (redV[w] > bv || (redV[w] == bv && redI[w] < bi)) { bv = redV[w]; bi = redI[w]; }
    *outV = bv; *outI = bi;
  }
  __syncthreads();
}

// Radix bucket selection over NBINS-entry histogram.
// strict=true  (count):  pick b with SC(b) <  budget <= SC(b)+h[b]
// strict=false (weight): pick b with SW(b) <= budget <  SW(b)+h[b]
// If scaleByTotal, budget := budgetIn * (total histogram mass); this lets the
// level-0 weight select use top_p * Z without a separate Z pass (Z == total).
// Writes {bucket, budget - suffix_above} into outSlot[0..1].
__device__ __forceinline__ void radix_select(const float* hist, float budgetIn,
                                             bool strict, bool scaleByTotal,
                                             float* outSlot, float* scan, int tid) {
  if (tid == 0) { outSlot[0] = 0.0f; outSlot[1] = 0.0f; }  // rounding fallback
  float h0 = hist[2 * tid];
  float h1 = hist[2 * tid + 1];
  scan[tid] = h0 + h1;
  __syncthreads();
  // Inclusive suffix scan over the 1024 per-thread partials.
  for (int off = 1; off < BS; off <<= 1) {
    float v = (tid + off < BS) ? scan[tid + off] : 0.0f;
    __syncthreads();
    scan[tid] += v;
    __syncthreads();
  }
  const float budget = scaleByTotal ? budgetIn * scan[0] : budgetIn;
  float above = (tid + 1 < BS) ? scan[tid + 1] : 0.0f;  // mass in buckets > 2*tid+1
  float SC1 = above;
  float SC0 = above + h1;
  bool c1 = strict ? (SC1 < budget && SC1 + h1 >= budget)
                   : (SC1 <= budget && SC1 + h1 > budget);
  bool c0 = strict ? (SC0 < budget && SC0 + h0 >= budget)
                   : (SC0 <= budget && SC0 + h0 > budget);
  if (c1) { outSlot[0] = (float)(2 * tid + 1); outSlot[1] = budget - SC1; }
  if (c0) { outSlot[0] = (float)(2 * tid);     outSlot[1] = budget - SC0; }
  __syncthreads();
}

__global__ __launch_bounds__(BS)
void sampler_kernel(const float* __restrict__ logits,
                    const float* __restrict__ temps,
                    const int*   __restrict__ topk,
                    const float* __restrict__ topp,
                    const float* __restrict__ noise,
                    int* __restrict__ out,
                    int V) {
  __shared__ float s_cnt[NBINS];    // count histogram (exact: float adds of 1.0)
  __shared__ float s_wgt[NBINS];    // exp-weight histogram (top-p mass)
  __shared__ float s_scan[BS];
  __shared__ float s_redf[NW];
  __shared__ int   s_redi[NW];
  __shared__ float s_bf[2];
  __shared__ int   s_bi[2];
  __shared__ float s_selK[2];
  __shared__ float s_selP[2];
#if USE_ASYNC
  __shared__ float s_stage[2][CH];  // 32 KB double-buffer ring
#endif

  const int tid = threadIdx.x;
  const int row = blockIdx.x;
  const float* rowL = logits + (size_t)row * (size_t)V;
  const float* rowN = noise  + (size_t)row * (size_t)V;
  // Vector (b128) path only when rows stay 16B-aligned; else scalar covers all.
  const int V4 = ((V & 3) == 0) ? (V >> 2) : 0;
  const float4* rowL4 = (const float4*)rowL;

  // ---- Pass 1: row max + greedy argmax (HBM-cold pass) ----
  // global_prefetch_b8 runs 8 chunks (~128 KB) ahead to pull lines into L2;
  // the async global->LDS ring then drains L2 at short latency.
  float bm = -INFINITY;
  int   bi = 0x7fffffff;
#if USE_ASYNC
  {
    const int fullCh = V / CH;
    if (fullCh > 0) {
#pragma unroll
      for (int j = 0; j < 4; ++j)
        async_copy_b32(rowL + j * BS + tid, &s_stage[0][j * BS + tid]);
    }
    for (int c = 0; c < fullCh; ++c) {
      if (c + 8 < fullCh)
        __builtin_prefetch(rowL + (size_t)(c + 8) * CH + (tid << 2), 0, 0);
      if (c + 1 < fullCh) {
#pragma unroll
        for (int j = 0; j < 4; ++j)
          async_copy_b32(rowL + (size_t)(c + 1) * CH + j * BS + tid,
                         &s_stage[(c + 1) & 1][j * BS + tid]);
        wait_async_le4();   // chunk c complete (in-order), chunk c+1 in flight
      } else {
        wait_async_le0();
      }
#pragma unroll
      for (int j = 0; j < 4; ++j) {          // each thread reads only its slots
        float l = s_stage[c & 1][j * BS + tid];
        int   i = c * CH + j * BS + tid;
        if (l > bm || (l == bm && i < bi)) { bm = l; bi = i; }
      }
    }
    for (int i = fullCh * CH + tid; i < V; i += BS) {
      float l = rowL[i];
      if (l > bm || (l == bm && i < bi)) { bm = l; bi = i; }
    }
    __syncthreads();
  }
#else
  for (int i4 = tid; i4 < V4; i4 += BS) {
    if (i4 + 8 * BS < V4) __builtin_prefetch(rowL4 + i4 + 8 * BS, 0, 0);
    float4 l4 = rowL4[i4];
    const float lv[4] = {l4.x, l4.y, l4.z, l4.w};
#pragma unroll
    for (int c = 0; c < 4; ++c) {
      int i = 4 * i4 + c;
      if (lv[c] > bm || (lv[c] == bm && i < bi)) { bm = lv[c]; bi = i; }
    }
  }
  for (int i = 4 * V4 + tid; i < V; i += BS) {
    float l = rowL[i];
    if (l > bm || (l == bm && i < bi)) { bm = l; bi = i; }
  }
#endif
  block_argmax(bm, bi, tid, s_redf, s_redi, &s_bf[0], &s_bi[0]);
  const float rowMax = s_bf[0];
  const int   greedy = s_bi[0];
  __syncthreads();

  // Per-row parameters (uniform scalar loads).
  const int   kRaw   = topk[row];
  const float kEff   = (kRaw <= 0 || kRaw > V) ? (float)V : (float)kRaw;
  const float temp   = temps[row];
  const float rowTopP = topp[row];
  float kRem = kEff;   // count budget (top-k select)
  float pRem = 0.0f;   // mass budget; level 0 derives it as top_p * Z in-select

  // ---- Passes 2-4: dual radix select (exact k-th largest logit and exact
  //      top-p boundary logit) over 32-bit keys, digits 11/11/10. L2-hot. ----
  unsigned prefK = 0u, prefP = 0u;
  for (int lvl = 0; lvl < 3; ++lvl) {
    const int      shift   = (lvl == 0) ? 21 : (lvl == 1) ? 10 : 0;
    const int      bits    = (lvl == 2) ? 10 : 11;
    const unsigned mask    = (1u << bits) - 1u;
    const int      hiShift = shift + bits;  // 32 (unused), 21, 10

    s_cnt[tid] = 0.0f; s_cnt[tid + BS] = 0.0f;
    s_wgt[tid] = 0.0f; s_wgt[tid + BS] = 0.0f;
    __syncthreads();

    for (int i4 = tid; i4 < V4; i4 += BS) {
      float4 l4 = rowL4[i4];
      const float lv[4] = {l4.x, l4.y, l4.z, l4.w};
#pragma unroll
      for (int c = 0; c < 4; ++c) {
        unsigned key   = key_of(lv[c]);
        unsigned hi    = (lvl == 0) ? 0u : (key >> hiShift);
        unsigned digit = (key >> shift) & mask;
        if (hi == prefK) atomicAdd(&s_cnt[digit], 1.0f);          // ds_add_f32
        if (hi == prefP) atomicAdd(&s_wgt[digit], expf(lv[c] - rowMax));
      }
    }
    for (int i = 4 * V4 + tid; i < V; i += BS) {
      float    l     = rowL[i];
      unsigned key   = key_of(l);
      unsigned hi    = (lvl == 0) ? 0u : (key >> hiShift);
      unsigned digit = (key >> shift) & mask;
      if (hi == prefK) atomicAdd(&s_cnt[digit], 1.0f);
      if (hi == prefP) atomicAdd(&s_wgt[digit], expf(l - rowMax));
    }
    __syncthreads();

    radix_select(s_cnt, kRem, /*strict=*/true, /*scaleByTotal=*/false,
                 s_selK, s_scan, tid);
    radix_select(s_wgt, (lvl == 0) ? rowTopP : pRem, /*strict=*/false,
                 /*scaleByTotal=*/(lvl == 0), s_selP, s_scan, tid);

    prefK = (prefK << bits) | (unsigned)s_selK[0];
    prefP = (prefP << bits) | (unsigned)s_selP[0];
    kRem  = s_selK[1];
    pRem  = s_selP[1];
    __syncthreads();
  }
  // Keep token iff key(l) >= max(t_k, t_p): matches (logits < thr -> -inf)
  // for top-k and prefix-mass(l) <= top_p for top-p.
  const unsigned thr = (prefK > prefP) ? prefK : prefP;

  // ---- Pass 5: sample = argmax over kept of l/temp - log(noise) ----
  const float invT = 1.0f / fmaxf(temp, 1e-5f);
  float bestS = -INFINITY;
  int   bestI = 0x7fffffff;
  for (int i4 = tid; i4 < V4; i4 += BS) {
    float4 l4 = rowL4[i4];
    const float lv[4] = {l4.x, l4.y, l4.z, l4.w};
#pragma unroll
    for (int c = 0; c < 4; ++c) {
      if (key_of(lv[c]) >= thr) {            // sparse: noise read only if kept
        int   i  = 4 * i4 + c;
        float n  = fmaxf(rowN[i], 1e-10f);
        float sc = lv[c] * invT - logf(n);
        if (sc > bestS || (sc == bestS && i < bestI)) { bestS = sc; bestI = i; }
      }
    }
  }
  for (int i = 4 * V4 + tid; i < V; i += BS) {
    float l = rowL[i];
    if (key_of(l) >= thr) {
      float n  = fmaxf(rowN[i], 1e-10f);
      float sc = l * invT - logf(n);
      if (sc > bestS || (sc == bestS && i < bestI)) { bestS = sc; bestI = i; }
    }
  }
  block_argmax(bestS, bestI, tid, s_redf, s_redi, &s_bf[1], &s_bi[1]);
  if (tid == 0) out[row] = (temp < 1e-5f) ? greedy : s_bi[1];
}

extern "C" void kernel_launch(void* const* d_in, const int* in_sizes, int n_in,
                              void* d_out, int out_size, void* d_ws, size_t ws_size,
                              hipStream_t stream) {
  (void)n_in; (void)out_size; (void)d_ws; (void)ws_size;
  const float* logits = (const float*)d_in[0];
  const float* temps  = (const float*)d_in[1];
  const int*   topk   = (const int*)d_in[2];
  const float* topp   = (const float*)d_in[3];
  const float* noise  = (const float*)d_in[4];
  const int B = in_sizes[1];
  if (B <= 0) return;
  const int V = (int)(in_sizes[0] / B);
  if (V <= 0) return;
  int* out = (int*)d_out;  // reference output dtype is int32 (token indices)
  hipLaunchKernelGGL(sampler_kernel, dim3(B), dim3(BS), 0, stream,
                     logits, temps, topk, topp, noise, out, V);
}